// HebbianLearning_9852654977807
// MI455X (gfx1250) — compile-verified
//
#include <hip/hip_runtime.h>

typedef __attribute__((ext_vector_type(2))) float v2f;
typedef __attribute__((ext_vector_type(8))) float v8f;
typedef __attribute__((ext_vector_type(4))) unsigned int u32x4;
typedef __attribute__((ext_vector_type(4))) int i32x4;
typedef __attribute__((ext_vector_type(8))) int i32x8;

#define B_   8
#define C_   16
#define H_   34
#define W_   34
#define KH   5
#define KW   5
#define HO   30
#define WO   30
#define KOUT 96
#define NTOT (B_ * HO * WO)   /* 7200 */
#define D_   (C_ * KH * KW)   /* 400  */
#define NT   6                /* KOUT/16 column tiles   */
#define SPLITK 8
#define KCHUNK (NTOT / SPLITK) /* 900 = exactly one batch */
#define WCOLS (D_ + KOUT)      /* 496 */

__device__ __forceinline__ v8f wmma4(v2f a, v2f b, v8f c) {
  // D = A(16x4 f32) * B(4x16 f32) + C(16x16 f32)
  return __builtin_amdgcn_wmma_f32_16x16x4_f32(
      /*neg_a=*/false, a, /*neg_b=*/false, b,
      /*c_mod=*/(short)0, c, /*reuse_a=*/false, /*reuse_b=*/false);
}

// ---------------------------------------------------------------------------
// Kernel 1: forward GEMM  S[n,k] = sum_d Xunf[n,d] * W[k,d]
//  - 8 waves/block, one 16-row tile per wave, all 6 output-channel tiles per
//    wave (A-fragment reused 6x per gather).
//  - W (96x400 f32) staged into dynamic LDS by the Tensor Data Mover (wave 0),
//    fenced with s_wait_tensorcnt + barrier.
//  - unfold offsets come from a per-block LDS table (no divides in hot loop).
//  - B-fragments batched ahead of the WMMA chain; next A-fragment prefetched
//    under the current step's WMMAs (branchless index clamp).
// ---------------------------------------------------------------------------
__global__ __launch_bounds__(256)
void fwd_gemm(const float* __restrict__ x, const float* __restrict__ w,
              float* __restrict__ out, float* __restrict__ s)
{
  __shared__ int xoffTab[D_];
  extern __shared__ float wlds[];          // KOUT*D_ floats via TDM

  const int tid = threadIdx.x;

  // unfold offset table: d -> c*H*W + i*W + j
  for (int d = tid; d < D_; d += 256) {
    const int c = d / 25, r = d - c * 25, i = r / 5, j = r - i * 5;
    xoffTab[d] = c * (H_ * W_) + i * W_ + j;
  }

  if (tid < 32) {
    // TDM descriptor: 2-D tensor, 400 wide x 96 rows, tile == whole tensor.
    const unsigned long long wa = (unsigned long long)(uintptr_t)w;
    u32x4 g0;
    g0[0] = 1u;                                            // count=1, user mode
    g0[1] = (unsigned)__builtin_amdgcn_groupstaticsize();  // lds_addr: dyn-LDS base
    g0[2] = (unsigned)(wa & 0xffffffffu);                  // global_addr[31:0]
    g0[3] = (unsigned)((wa >> 32) & 0x01ffffffu) | (2u << 30); // addr[56:32] | type=2
    i32x8 g1;
    g1[0] = (int)(2u << 16);                // data_size = 4 bytes
    g1[1] = (int)(((unsigned)D_) << 16);    // tensor_dim0[15:0] = 400
    g1[2] = (int)(((unsigned)KOUT) << 16);  // tensor_dim0[31:16]=0 | tensor_dim1=96
    g1[3] = (int)(((unsigned)D_) << 16);    // tensor_dim1 hi=0 | tile_dim0 = 400
    g1[4] = (int)KOUT;                      // tile_dim1 = 96 | tile_dim2 = 0
    g1[5] = (int)D_;                        // tensor_dim0_stride = 400 (lo32)
    g1[6] = 0;                              // stride0 hi | stride1 lo
    g1[7] = 0;
    i32x4 gz4 = {0, 0, 0, 0};
    i32x8 gz8 = {0, 0, 0, 0, 0, 0, 0, 0};
    __builtin_amdgcn_tensor_load_to_lds(g0, g1, gz4, gz4, gz8, 0);
    __builtin_amdgcn_s_wait_tensorcnt(0);
  }
  __syncthreads();

  const int wave = tid >> 5;
  const int lane = tid & 31;
  const int m    = lane & 15;
  const int half = lane >> 4;
  const int rt   = blockIdx.x * 8 + wave;      // 16-row tile index
  if (rt >= NTOT / 16) return;

  const int row = rt * 16 + m;
  const int bb  = row / (HO * WO);
  const int rem = row - bb * (HO * WO);
  const int oh  = rem / WO;
  const int ow  = rem - oh * WO;
  const int xbase = bb * (C_ * H_ * W_) + oh * W_ + ow;

  v8f acc[NT] = {};

  // A-fragment pipeline prologue
  int d0 = half * 2;
  v2f a;
  a.x = x[xbase + xoffTab[d0]];
  a.y = x[xbase + xoffTab[d0 + 1]];

  for (int kk = 0; kk < D_; kk += 4) {
    // batch all 6 B-fragments from LDS ahead of the WMMA chain
    v2f b[NT];
#pragma unroll
    for (int t = 0; t < NT; ++t) {
      const float* wp = &wlds[(t * 16 + m) * D_ + d0];
      b[t].x = wp[0];
      b[t].y = wp[1];
    }
    // prefetch next A-fragment (branchless clamp keeps the access in-range)
    int dn = d0 + 4;
    const int dnc = (dn < D_) ? dn : 0;
    v2f an;
    an.x = x[xbase + xoffTab[dnc]];
    an.y = x[xbase + xoffTab[dnc + 1]];
#pragma unroll
    for (int t = 0; t < NT; ++t)
      acc[t] = wmma4(a, b[t], acc[t]);
    a  = an;
    d0 = dn;
  }

#pragma unroll
  for (int t = 0; t < NT; ++t) {
#pragma unroll
    for (int v = 0; v < 8; ++v) {
      const int rr = rt * 16 + (half ? v + 8 : v);
      const int cc = t * 16 + m;
      const int b2 = rr / (HO * WO);
      const int rm = rr - b2 * (HO * WO);
      const int o2 = rm / WO;
      const int p2 = rm - o2 * WO;
      const float val = acc[t][v];
      out[((b2 * KOUT + cc) * HO + o2) * WO + p2] = val;
      s[rr * KOUT + cc] = val;
    }
  }
}

// ---------------------------------------------------------------------------
// Kernel 2: c[n,k] = s*|s| ;  r_sum[k] = sum_n |s[n,k]| (zero-guarded)
// deterministic LDS tree reduction (graph replays stay bit-stable).
// ---------------------------------------------------------------------------
__global__ __launch_bounds__(256)
void cr_kernel(const float* __restrict__ s, float* __restrict__ cbuf,
               float* __restrict__ rsum)
{
  const int k   = blockIdx.x;
  const int tid = threadIdx.x;
  __shared__ float sm[256];

  float partial = 0.f;
  for (int n = tid; n < NTOT; n += 256) {
    const float v  = s[n * KOUT + k];
    const float av = fabsf(v);
    cbuf[n * KOUT + k] = v * av;
    partial += av;
  }
  sm[tid] = partial;
  __syncthreads();
  for (int st = 128; st > 0; st >>= 1) {
    if (tid < st) sm[tid] += sm[tid + st];
    __syncthreads();
  }
  if (tid == 0) {
    const float t = sm[0];
    rsum[k] = t + (t == 0.f ? 1.f : 0.f);
  }
}

// ---------------------------------------------------------------------------
// Kernel 3: split-K partials of  [A | M] = C^T (96 x 7200) * [Xunf | S]
//  - one wave per (column tile, K-chunk); all 6 row tiles per wave so the
//    expensive x-unfold B-fragment feeds 6 WMMAs.
//  - n -> x-base offsets precomputed into LDS (no divides in hot loop).
//  - A-fragments batched ahead of the WMMA chain; next B-fragment prefetched
//    under the current step's WMMAs.
// ---------------------------------------------------------------------------
__global__ __launch_bounds__(32)
void upd_gemm(const float* __restrict__ x, const float* __restrict__ s,
              const float* __restrict__ cbuf, float* __restrict__ part)
{
  __shared__ int ntab[KCHUNK];
  const int lane = threadIdx.x;
  const int m    = lane & 15;
  const int half = lane >> 4;
  const int NCT  = WCOLS / 16;                    // 31 column tiles
  const int tn   = (blockIdx.x % NCT) * 16;
  const int chunk = blockIdx.x / NCT;
  const int nbase = chunk * KCHUNK;

  for (int q = lane; q < KCHUNK; q += 32) {
    const int n = nbase + q;
    const int b = n / (HO * WO), r = n - b * (HO * WO);
    const int o = r / WO, p = r - o * WO;
    ntab[q] = b * (C_ * H_ * W_) + o * W_ + p;
  }
  __syncthreads();

  const int nc   = tn + m;
  const bool isx = (nc < D_);                     // wave-uniform (no straddle)
  int xoff = 0;
  if (isx) {
    const int cx = nc / 25, r = nc - cx * 25, ix = r / 5, jx = r - ix * 5;
    xoff = cx * (H_ * W_) + ix * W_ + jx;
  }
  const int scol = nc - D_;

  auto loadB = [&](int q0) -> v2f {
    v2f b;
    if (isx) {
      b.x = x[ntab[q0] + xoff];
      b.y = x[ntab[q0 + 1] + xoff];
    } else {
      const int n0 = nbase + q0;
      b.x = s[n0 * KOUT + scol];
      b.y = s[(n0 + 1) * KOUT + scol];
    }
    return b;
  };

  v8f acc[NT] = {};
  v2f b = loadB(half * 2);                        // pipeline prologue

  for (int q = 0; q < KCHUNK; q += 4) {
    const int q0 = q + half * 2;
    // batch all 6 A-fragments (cbuf rows) ahead of the WMMA chain
    v2f a[NT];
    const float* c0 = cbuf + (size_t)(nbase + q0) * KOUT + m;
#pragma unroll
    for (int t = 0; t < NT; ++t) {
      a[t].x = c0[t * 16];
      a[t].y = c0[KOUT + t * 16];
    }
    // prefetch next B-fragment (branchless in-range clamp)
    int qn = q0 + 4;
    if (qn + 1 >= KCHUNK) qn = half * 2;
    const v2f bn = loadB(qn);
#pragma unroll
    for (int t = 0; t < NT; ++t)
      acc[t] = wmma4(a[t], b, acc[t]);
    b = bn;
  }

#pragma unroll
  for (int t = 0; t < NT; ++t) {
#pragma unroll
    for (int v = 0; v < 8; ++v) {
      const int rr = t * 16 + (half ? v + 8 : v);
      const int cc = tn + m;
      part[((size_t)chunk * KOUT + rr) * WCOLS + cc] = acc[t][v];
    }
  }
}

// ---------------------------------------------------------------------------
// Kernel 4: reduce split-K partials -> A (96x400), M (96x96)
// ---------------------------------------------------------------------------
__global__ __launch_bounds__(256)
void reduce_upd(const float* __restrict__ part, float* __restrict__ A2,
                float* __restrict__ Mm)
{
  const int idx = blockIdx.x * 256 + threadIdx.x;
  const int TOT = KOUT * WCOLS;
  if (idx >= TOT) return;
  float acc = 0.f;
#pragma unroll
  for (int c = 0; c < SPLITK; ++c) acc += part[(size_t)c * TOT + idx];
  const int rr = idx / WCOLS, cc = idx - rr * WCOLS;
  if (cc < D_) A2[rr * D_ + cc] = acc;
  else         Mm[rr * KOUT + (cc - D_)] = acc;
}

// ---------------------------------------------------------------------------
// Kernel 5: dw[k,d] = (A[k,d] - sum_{k'<=k} M[k,k'] * W[k',d]) / r_sum[k]
// ---------------------------------------------------------------------------
__global__ __launch_bounds__(256)
void dw_kernel(const float* __restrict__ w, const float* __restrict__ A2,
               const float* __restrict__ Mm, const float* __restrict__ rsum,
               float* __restrict__ dw)
{
  const int idx = blockIdx.x * 256 + threadIdx.x;
  if (idx >= KOUT * D_) return;
  const int k = idx / D_;
  const int d = idx - k * D_;

  float acc = A2[idx];
  const float* mrow = Mm + k * KOUT;
  for (int kp = 0; kp <= k; ++kp)
    acc -= mrow[kp] * w[kp * D_ + d];
  dw[idx] = acc / rsum[k];
}

// ---------------------------------------------------------------------------
extern "C" void kernel_launch(void* const* d_in, const int* in_sizes, int n_in,
                              void* d_out, int out_size, void* d_ws, size_t ws_size,
                              hipStream_t stream)
{
  const float* x = (const float*)d_in[0];   // (8,16,34,34)
  const float* w = (const float*)d_in[1];   // (96,16,5,5)
  // d_in[2] = label : unused (ff_activate disabled in reference)

  float* out = (float*)d_out;               // (8,96,30,30) = 691200
  float* dw  = out + B_ * KOUT * HO * WO;   // (96,16,5,5)  = 38400

  float* ws   = (float*)d_ws;
  float* s    = ws;                          // 691200 f
  float* cbuf = s + NTOT * KOUT;             // 691200 f
  float* part = cbuf + NTOT * KOUT;          // SPLITK*96*496 = 380928 f
  float* A2   = part + SPLITK * KOUT * WCOLS;// 38400 f
  float* Mm   = A2 + KOUT * D_;              // 9216 f
  float* rsum = Mm + KOUT * KOUT;            // 96 f   (~7.2 MB total)

  const int rowTiles = NTOT / 16;            // 450
  fwd_gemm<<<(rowTiles + 7) / 8, 256, KOUT * D_ * sizeof(float), stream>>>(x, w, out, s);
  cr_kernel<<<KOUT, 256, 0, stream>>>(s, cbuf, rsum);
  upd_gemm<<<(WCOLS / 16) * SPLITK, 32, 0, stream>>>(x, s, cbuf, part);
  reduce_upd<<<(KOUT * WCOLS + 255) / 256, 256, 0, stream>>>(part, A2, Mm);
  dw_kernel<<<(KOUT * D_ + 255) / 256, 256, 0, stream>>>(w, A2, Mm, rsum, dw);
}